// AngleFreqEnhanceFPN_77369540870375
// MI455X (gfx1250) — compile-verified
//
#include <hip/hip_runtime.h>
#include <math.h>

#define PI_F 3.14159265358979323846f
#define EPS_F 1e-8f
#define FPITCH 129   // 128 + 1 pad -> conflict-free row & column FFT phases on 64 banks

typedef __attribute__((ext_vector_type(2))) float v2f;
typedef __attribute__((ext_vector_type(8))) float v8f;

// ---------------------------------------------------------------------------
// Serial 128-point radix-2 DIT FFT over strided LDS data (one FFT per thread).
// sign = -1 forward, +1 inverse (unscaled).
// ---------------------------------------------------------------------------
__device__ inline void fft128(float* re, float* im, int stride, float sign) {
  #pragma unroll 1
  for (int i = 0; i < 128; ++i) {
    int j = (int)(__brev((unsigned)i) >> 25);   // 7-bit reversal
    if (j > i) {
      float tr = re[i * stride]; re[i * stride] = re[j * stride]; re[j * stride] = tr;
      float ti = im[i * stride]; im[i * stride] = im[j * stride]; im[j * stride] = ti;
    }
  }
  #pragma unroll 1
  for (int len = 2; len <= 128; len <<= 1) {
    float ang = sign * 2.0f * PI_F / (float)len;
    #pragma unroll 1
    for (int i = 0; i < 128; i += len) {
      #pragma unroll 1
      for (int j = 0; j < (len >> 1); ++j) {
        float s, c;
        __sincosf(ang * (float)j, &s, &c);
        int i0 = (i + j) * stride;
        int i1 = (i + j + (len >> 1)) * stride;
        float r1 = re[i1], m1 = im[i1];
        float tr = r1 * c - m1 * s;
        float ti = r1 * s + m1 * c;
        re[i1] = re[i0] - tr; im[i1] = im[i0] - ti;
        re[i0] += tr;         im[i0] += ti;
      }
    }
  }
}

// ---------------------------------------------------------------------------
// Forward 2D FFT (ortho) of one 128x128 slice + fftshift on store.
// One block per (b,c) slice, 128 threads, full slice in LDS.
// ---------------------------------------------------------------------------
__global__ __launch_bounds__(128) void fft_fwd_kernel(const float* __restrict__ x,
                                                      float* __restrict__ outRe,
                                                      float* __restrict__ outIm) {
  extern __shared__ float smem[];
  float* sre = smem;
  float* sim = smem + 128 * FPITCH;
  int slice = blockIdx.x;
  const float* src = x + (size_t)slice * 16384;
  for (int i = threadIdx.x; i < 16384; i += 128) {
    int r = i >> 7, c = i & 127;
    sre[r * FPITCH + c] = src[i];
    sim[r * FPITCH + c] = 0.0f;
  }
  __syncthreads();
  int t = threadIdx.x;
  fft128(&sre[t * FPITCH], &sim[t * FPITCH], 1, -1.0f);   // row t
  __syncthreads();
  fft128(&sre[t], &sim[t], FPITCH, -1.0f);                // col t
  __syncthreads();
  float* dre = outRe + (size_t)slice * 16384;
  float* dim_ = outIm + (size_t)slice * 16384;
  const float sc = 1.0f / 128.0f;                          // ortho
  for (int i = threadIdx.x; i < 16384; i += 128) {
    int r = i >> 7, c = i & 127;
    int rs = (r + 64) & 127, cs = (c + 64) & 127;          // fftshift
    dre[rs * 128 + cs] = sre[r * FPITCH + c] * sc;
    dim_[rs * 128 + cs] = sim[r * FPITCH + c] * sc;
  }
}

// ---------------------------------------------------------------------------
// Center-patch magnitude -> MLP(49->32->3) -> anisotropic Gaussian 7x7 kernel.
// One block per batch sample.
// ---------------------------------------------------------------------------
__global__ __launch_bounds__(64) void param_kernel(const float* __restrict__ fftRe,
                                                   const float* __restrict__ fftIm,
                                                   const float* __restrict__ W1,
                                                   const float* __restrict__ b1,
                                                   const float* __restrict__ W2,
                                                   const float* __restrict__ b2,
                                                   float* __restrict__ kbuf) {
  __shared__ float center[49];
  __shared__ float hid[32];
  __shared__ float prm[3];
  __shared__ float kv[49];
  __shared__ float ksum;
  int b = blockIdx.x;
  for (int p = threadIdx.x; p < 49; p += 64) {
    int y = 61 + p / 7, x = 61 + p % 7;                    // h0 = 128/2 - 3
    float s = 0.0f;
    for (int c = 0; c < 256; ++c) {
      size_t idx = (((size_t)b * 256 + c) * 128 + y) * 128 + x;
      float re = fftRe[idx], im = fftIm[idx];
      s += sqrtf(re * re + im * im);
    }
    center[p] = s * (1.0f / 256.0f);                       // mean over channels
  }
  __syncthreads();
  if (threadIdx.x < 32) {
    float a = b1[threadIdx.x];
    for (int j = 0; j < 49; ++j) a = fmaf(W1[threadIdx.x * 49 + j], center[j], a);
    hid[threadIdx.x] = fmaxf(a, 0.0f);
  }
  __syncthreads();
  if (threadIdx.x < 3) {
    float a = b2[threadIdx.x];
    for (int j = 0; j < 32; ++j) a = fmaf(W2[threadIdx.x * 32 + j], hid[j], a);
    prm[threadIdx.x] = a;
  }
  __syncthreads();
  float theta = atan2f(prm[0], prm[1]) * 0.5f + PI_F * 0.5f;
  float lam1 = expf(prm[2]);
  float lam2 = 1.0f / (lam1 + EPS_F);
  float ct = cosf(theta), st = sinf(theta);
  if (threadIdx.x < 49) {
    int p = threadIdx.x;
    float yv = (float)(p / 7 - 3), xv = (float)(p % 7 - 3);
    float xr = xv * ct + yv * st;
    float yr = -xv * st + yv * ct;
    kv[p] = expf(-(xr * xr / (2.0f * lam1 * lam1) + yr * yr / (2.0f * lam2 * lam2)));
  }
  __syncthreads();
  if (threadIdx.x == 0) {
    float s = 0.0f;
    for (int p = 0; p < 49; ++p) s += kv[p];
    ksum = s + EPS_F;
  }
  __syncthreads();
  if (threadIdx.x < 49) kbuf[b * 49 + threadIdx.x] = kv[threadIdx.x] / ksum;
}

// ---------------------------------------------------------------------------
// Depthwise 7x7 conv (zero pad 3) on shifted spectra; re & im in one launch, one pass each.
// Block = 16x16 output tile, 22x22 halo tile staged in LDS.
// ---------------------------------------------------------------------------
__global__ __launch_bounds__(256) void dwconv_kernel(const float* __restrict__ inRe,
                                                     const float* __restrict__ inIm,
                                                     const float* __restrict__ kbuf,
                                                     float* __restrict__ outRe,
                                                     float* __restrict__ outIm) {
  __shared__ float sk[49];
  __shared__ float tile[22 * 22];
  int slice = blockIdx.z;            // b*256 + c
  int b = slice >> 8;
  if (threadIdx.x < 49) sk[threadIdx.x] = kbuf[b * 49 + threadIdx.x];
  int tx = threadIdx.x & 15, ty = threadIdx.x >> 4;
  int x0 = blockIdx.x * 16, y0 = blockIdx.y * 16;
  const float* srcs[2] = {inRe, inIm};
  float* dsts[2] = {outRe, outIm};
  #pragma unroll 1
  for (int pass = 0; pass < 2; ++pass) {
    const float* src = srcs[pass] + (size_t)slice * 16384;
    __syncthreads();
    for (int i = threadIdx.x; i < 484; i += 256) {
      int yy = y0 - 3 + i / 22, xx = x0 - 3 + i % 22;
      float v = 0.0f;
      if (yy >= 0 && yy < 128 && xx >= 0 && xx < 128) v = src[yy * 128 + xx];
      tile[i] = v;
    }
    __syncthreads();
    float acc = 0.0f;
    #pragma unroll
    for (int dy = 0; dy < 7; ++dy)
      #pragma unroll
      for (int dx = 0; dx < 7; ++dx)
        acc = fmaf(tile[(ty + dy) * 22 + tx + dx], sk[dy * 7 + dx], acc);
    dsts[pass][(size_t)slice * 16384 + (size_t)(y0 + ty) * 128 + x0 + tx] = acc;
  }
}

// ---------------------------------------------------------------------------
// refine 1x1 conv as WMMA f32 GEMM:  Y[o,pix] = sum_c W[o,c] * X[c,pix] + b[o]
// per batch: M=256(o), K=256(c), N=16384(pix).  V_WMMA_F32_16X16X4_F32.
// Block = 4 waves; each wave owns a 16(M) x 64(N) strip (4 accumulators).
// W row-block (16x256) staged in LDS and shared by all waves.
//   A 16x4 f32 frag:  v0 = K{0|2}, v1 = K{1|3}; lanes 0-15 -> M, 16-31 -> M (K+2)
//   B 4x16 f32 frag:  v0 = rows K{0|2}, v1 = rows K{1|3}; lane%16 -> N
//   C/D 16x16:        vgpr v -> M = v (+8 upper lanes), lane%16 -> N
// ---------------------------------------------------------------------------
__global__ __launch_bounds__(128) void refine_wmma_kernel(const float* __restrict__ X,
                                                          const float* __restrict__ Wm,
                                                          const float* __restrict__ bias,
                                                          float* __restrict__ Y) {
  __shared__ float sW[16 * 256];     // 16 KB: W rows for this M tile
  int b = blockIdx.z, mt = blockIdx.y;
  for (int i = threadIdx.x; i < 16 * 256; i += 128) {
    int m = i >> 8, k = i & 255;
    sW[i] = Wm[(size_t)(mt * 16 + m) * 256 + k];
  }
  __syncthreads();

  int wave = threadIdx.x >> 5;
  int lane = threadIdx.x & 31;
  int l16 = lane & 15;               // A: M row within tile;  B/D: N col within tile
  int khalf = (lane >> 4) << 1;      // 0 for lanes 0-15, 2 for lanes 16-31
  int pix0 = blockIdx.x * 256 + wave * 64;

  const float* xb = X + (size_t)b * 256 * 16384;
  v8f acc0 = {}, acc1 = {}, acc2 = {}, acc3 = {};

  #pragma unroll 4
  for (int k0 = 0; k0 < 256; k0 += 4) {
    v2f a = *(const v2f*)(&sW[l16 * 256 + k0 + khalf]);          // K and K+1
    const float* brow = xb + (size_t)(k0 + khalf) * 16384 + pix0 + l16;
    __builtin_prefetch(brow + 4 * 16384, 0, 1);                  // stream next K rows
    v2f b0, b1, b2, b3;
    b0.x = brow[0];  b0.y = brow[16384];
    b1.x = brow[16]; b1.y = brow[16 + 16384];
    b2.x = brow[32]; b2.y = brow[32 + 16384];
    b3.x = brow[48]; b3.y = brow[48 + 16384];
    acc0 = __builtin_amdgcn_wmma_f32_16x16x4_f32(false, a, false, b0, (short)0, acc0, false, false);
    acc1 = __builtin_amdgcn_wmma_f32_16x16x4_f32(false, a, false, b1, (short)0, acc1, false, false);
    acc2 = __builtin_amdgcn_wmma_f32_16x16x4_f32(false, a, false, b2, (short)0, acc2, false, false);
    acc3 = __builtin_amdgcn_wmma_f32_16x16x4_f32(false, a, false, b3, (short)0, acc3, false, false);
  }

  int mbase = mt * 16 + ((lane >> 4) << 3);   // +8 for upper half-wave
  float* yb = Y + (size_t)b * 256 * 16384;
  v8f accs[4] = {acc0, acc1, acc2, acc3};
  #pragma unroll
  for (int nt = 0; nt < 4; ++nt) {
    size_t pix = (size_t)pix0 + nt * 16 + l16;
    #pragma unroll
    for (int v = 0; v < 8; ++v) {
      int o = mbase + v;
      yb[(size_t)o * 16384 + pix] = accs[nt][v] + bias[o];
    }
  }
}

// ---------------------------------------------------------------------------
// ifftshift -> inverse 2D FFT (ortho) -> real part + fused bilinear upsample.
// ---------------------------------------------------------------------------
__global__ __launch_bounds__(128) void ifft_fuse_kernel(const float* __restrict__ inRe,
                                                        const float* __restrict__ inIm,
                                                        const float* __restrict__ xh,
                                                        float* __restrict__ out) {
  extern __shared__ float smem[];
  float* sre = smem;
  float* sim = smem + 128 * FPITCH;
  int slice = blockIdx.x;
  const float* gre = inRe + (size_t)slice * 16384;
  const float* gim = inIm + (size_t)slice * 16384;
  for (int i = threadIdx.x; i < 16384; i += 128) {
    int r = i >> 7, c = i & 127;
    int rs = (r + 64) & 127, cs = (c + 64) & 127;          // ifftshift (N even)
    sre[r * FPITCH + c] = gre[rs * 128 + cs];
    sim[r * FPITCH + c] = gim[rs * 128 + cs];
  }
  __syncthreads();
  int t = threadIdx.x;
  fft128(&sre[t * FPITCH], &sim[t * FPITCH], 1, +1.0f);    // inverse rows
  __syncthreads();
  fft128(&sre[t], &sim[t], FPITCH, +1.0f);                 // inverse cols
  __syncthreads();
  const float* hb = xh + (size_t)slice * 4096;             // 64x64 high-level slice
  float* ob = out + (size_t)slice * 16384;
  const float sc = 1.0f / 128.0f;                          // ortho inverse
  for (int i = threadIdx.x; i < 16384; i += 128) {
    int r = i >> 7, c = i & 127;
    float sy = 0.5f * (float)r - 0.25f;                    // align_corners=False
    float sx = 0.5f * (float)c - 0.25f;
    int y0 = (int)floorf(sy), x0 = (int)floorf(sx);
    float fy = sy - (float)y0, fx = sx - (float)x0;
    int y0c = y0 < 0 ? 0 : (y0 > 63 ? 63 : y0);
    int y1c = (y0 + 1) < 0 ? 0 : ((y0 + 1) > 63 ? 63 : (y0 + 1));
    int x0c = x0 < 0 ? 0 : (x0 > 63 ? 63 : x0);
    int x1c = (x0 + 1) < 0 ? 0 : ((x0 + 1) > 63 ? 63 : (x0 + 1));
    float v00 = hb[y0c * 64 + x0c], v01 = hb[y0c * 64 + x1c];
    float v10 = hb[y1c * 64 + x0c], v11 = hb[y1c * 64 + x1c];
    float up = v00 * (1.0f - fy) * (1.0f - fx) + v01 * (1.0f - fy) * fx
             + v10 * fy * (1.0f - fx) + v11 * fy * fx;
    ob[i] = sre[r * FPITCH + c] * sc + up;
  }
}

// ---------------------------------------------------------------------------
extern "C" void kernel_launch(void* const* d_in, const int* in_sizes, int n_in,
                              void* d_out, int out_size, void* d_ws, size_t ws_size,
                              hipStream_t stream) {
  const float* x_high = (const float*)d_in[0];   // (8,256,64,64)
  const float* x_low  = (const float*)d_in[1];   // (8,256,128,128)
  const float* W1     = (const float*)d_in[2];   // (32,49)
  const float* b1     = (const float*)d_in[3];   // (32)
  const float* W2     = (const float*)d_in[4];   // (3,32)
  const float* b2     = (const float*)d_in[5];   // (3)
  const float* rW     = (const float*)d_in[6];   // (256,256)
  const float* rb     = (const float*)d_in[7];   // (256)
  float* out = (float*)d_out;

  const size_t NTOT = (size_t)8 * 256 * 128 * 128;   // 33,554,432 elems / buffer
  float* fftRe  = (float*)d_ws;
  float* fftIm  = fftRe + NTOT;
  float* convRe = fftIm + NTOT;
  float* convIm = convRe + NTOT;
  float* kbuf   = convIm + NTOT;                     // 8*49 coefficients

  size_t fftSmem = (size_t)2 * 128 * FPITCH * sizeof(float);  // 129 KB dyn LDS

  fft_fwd_kernel<<<2048, 128, fftSmem, stream>>>(x_low, fftRe, fftIm);
  param_kernel<<<8, 64, 0, stream>>>(fftRe, fftIm, W1, b1, W2, b2, kbuf);
  dwconv_kernel<<<dim3(8, 8, 2048), 256, 0, stream>>>(fftRe, fftIm, kbuf, convRe, convIm);
  refine_wmma_kernel<<<dim3(64, 16, 8), 128, 0, stream>>>(convRe, rW, rb, fftRe);
  refine_wmma_kernel<<<dim3(64, 16, 8), 128, 0, stream>>>(convIm, rW, rb, fftIm);
  ifft_fuse_kernel<<<2048, 128, fftSmem, stream>>>(fftRe, fftIm, x_high, out);
}